// MultiHeadAttention_87540023427471
// MI455X (gfx1250) — compile-verified
//
#include <hip/hip_runtime.h>

typedef __attribute__((ext_vector_type(2))) float v2f;
typedef __attribute__((ext_vector_type(8))) float v8f;

#define SEQ  102
#define HID  8
#define HDIM 4
#define SPAD 112      // 7*16 padded sequence (Q/K rows, score rows/cols)
#define VPAD 104      // 26*4: V rows needed by the chained-K WMMA phase
#define SROW 113      // odd stride -> conflict-free LDS score rows
#define NT   7        // 112/16 tiles per dim
#define NEGINF (-3.402823466e38f)

__global__ __launch_bounds__(256)
void mha_rope_kernel(const float* __restrict__ q_in,
                     const float* __restrict__ k_in,
                     const float* __restrict__ v_in,
                     const float* __restrict__ wq,
                     const float* __restrict__ wk,
                     const float* __restrict__ wv,
                     const float* __restrict__ wo,
                     float* __restrict__ out)
{
    __shared__ float sQ[2 * SPAD * HDIM];   // both heads, rotary+0.5 applied
    __shared__ float sK[2 * SPAD * HDIM];   // both heads, rotary applied
    __shared__ float sV[2 * VPAD * HDIM];   // both heads, raw
    __shared__ float sS[SPAD * SROW];       // score / prob tile (per head)
    __shared__ float sO[SEQ * HID];         // merged attention output
    __shared__ float sW[4 * 64];            // wq | wk | wv | wo

    const int tid  = threadIdx.x;
    const int b    = blockIdx.x;
    const int lane = tid & 31;
    const int wid  = tid >> 5;
    const int half = lane >> 4;     // 0: lanes 0-15, 1: lanes 16-31
    const int l16  = lane & 15;

    // Stage all four 8x8 weight matrices into LDS
    if (tid < 64)       sW[tid] = wq[tid];
    else if (tid < 128) sW[tid] = wk[tid - 64];
    else if (tid < 192) sW[tid] = wv[tid - 128];
    else                sW[tid] = wo[tid - 192];

    const float* xq = q_in + (size_t)b * SEQ * HID;
    const float* xk = k_in + (size_t)b * SEQ * HID;
    const float* xv = v_in + (size_t)b * SEQ * HID;
    __syncthreads();

    // ---- 1) projections + rotary for BOTH heads, single global pass ----
    // Reference rotary quirk: dim=2 -> inv_freq=[1]; cache[:, :s] broadcasts
    // [sin,cos] over the HEADS axis and sin_t==cos_t, so per (head hh, pos p):
    //   f = hh==0 ? sin(p) : cos(p);  rot(x) = [x0-x1, x1+x0, x2-x3, x3+x2]*f
    for (int idx = tid; idx < SEQ * 2; idx += 256) {
        int s  = idx >> 1;
        int hh = idx & 1;
        float xr[HID], kr[HID], vr[HID];
        *(float4*)&xr[0] = ((const float4*)(xq + s * HID))[0];
        *(float4*)&xr[4] = ((const float4*)(xq + s * HID))[1];
        *(float4*)&kr[0] = ((const float4*)(xk + s * HID))[0];
        *(float4*)&kr[4] = ((const float4*)(xk + s * HID))[1];
        *(float4*)&vr[0] = ((const float4*)(xv + s * HID))[0];
        *(float4*)&vr[4] = ((const float4*)(xv + s * HID))[1];

        float qh[HDIM], kh[HDIM], vh[HDIM];
        #pragma unroll
        for (int d = 0; d < HDIM; ++d) {
            int j = hh * HDIM + d;
            float aq = 0.f, ak = 0.f, av = 0.f;
            #pragma unroll
            for (int i = 0; i < HID; ++i) {
                aq += xr[i] * sW[0 * 64 + j * HID + i];
                ak += kr[i] * sW[1 * 64 + j * HID + i];
                av += vr[i] * sW[2 * 64 + j * HID + i];
            }
            qh[d] = aq; kh[d] = ak; vh[d] = av;
        }
        float f = (hh == 0) ? sinf((float)s) : cosf((float)s);
        float* pq = &sQ[(hh * SPAD + s) * HDIM];
        float* pk = &sK[(hh * SPAD + s) * HDIM];
        pq[0] = 0.5f * (qh[0] - qh[1]) * f;   // 0.5 = 1/sqrt(HEAD_DIM) folded
        pq[1] = 0.5f * (qh[1] + qh[0]) * f;
        pq[2] = 0.5f * (qh[2] - qh[3]) * f;
        pq[3] = 0.5f * (qh[3] + qh[2]) * f;
        pk[0] = (kh[0] - kh[1]) * f;
        pk[1] = (kh[1] + kh[0]) * f;
        pk[2] = (kh[2] - kh[3]) * f;
        pk[3] = (kh[3] + kh[2]) * f;
        float4 vst = { vh[0], vh[1], vh[2], vh[3] };
        *(float4*)&sV[(hh * VPAD + s) * HDIM] = vst;
    }
    // zero padded Q/K rows 102..111 and V rows 102..103, both heads
    for (int idx = tid; idx < (SPAD - SEQ) * HDIM * 2; idx += 256) {
        int hh = idx / ((SPAD - SEQ) * HDIM);
        int r  = idx % ((SPAD - SEQ) * HDIM);
        sQ[(hh * SPAD + SEQ) * HDIM + r] = 0.f;
        sK[(hh * SPAD + SEQ) * HDIM + r] = 0.f;
    }
    if (tid < (VPAD - SEQ) * HDIM * 2) {
        int hh = tid / ((VPAD - SEQ) * HDIM);
        int r  = tid % ((VPAD - SEQ) * HDIM);
        sV[(hh * VPAD + SEQ) * HDIM + r] = 0.f;
    }

    for (int h = 0; h < 2; ++h) {
        __syncthreads();  // phase1 done (h=0) / prev head's sS reads done (h=1)
        const float* hQ = &sQ[h * SPAD * HDIM];
        const float* hK = &sK[h * SPAD * HDIM];
        const float* hV = &sV[h * VPAD * HDIM];

        // ---- 2) scores = Qr @ Kr^T via V_WMMA_F32_16X16X4_F32 ----
        // A 16x4 f32: lanes 0-15 -> M=lane,K={0,1}; lanes 16-31 -> K={2,3}
        // B 4x16 f32: N=lane%16; lanes 0-15 -> K={0,1}; 16-31 -> K={2,3}
        for (int t = wid; t < NT * NT; t += 8) {
            int mt = t / NT, nt = t % NT;
            int m = mt * 16 + l16;
            int n = nt * 16 + l16;
            v2f a, bm;
            a.x  = hQ[m * HDIM + 2 * half];
            a.y  = hQ[m * HDIM + 2 * half + 1];
            bm.x = hK[n * HDIM + 2 * half];      // B[k][n] = Kr[n][k]
            bm.y = hK[n * HDIM + 2 * half + 1];
            v8f c = {};
            c = __builtin_amdgcn_wmma_f32_16x16x4_f32(
                    false, a, false, bm, (short)0, c, false, false);
            #pragma unroll
            for (int r = 0; r < 8; ++r)
                sS[(mt * 16 + r + 8 * half) * SROW + n] = c[r];
        }
        __syncthreads();

        // ---- 3) softmax: straight-line, 3 full + 1 guarded column chunk ----
        for (int r = wid; r < SEQ; r += 8) {
            int base = r * SROW;
            bool has3 = (lane + 96) < SEQ;   // lane < 6
            float x0 = sS[base + lane];
            float x1 = sS[base + lane + 32];
            float x2 = sS[base + lane + 64];
            float x3 = has3 ? sS[base + lane + 96] : NEGINF;
            float mx = fmaxf(fmaxf(x0, x1), fmaxf(x2, x3));
            #pragma unroll
            for (int off = 16; off >= 1; off >>= 1)
                mx = fmaxf(mx, __shfl_xor(mx, off, 32));
            float e0 = __expf(x0 - mx);
            float e1 = __expf(x1 - mx);
            float e2 = __expf(x2 - mx);
            float e3 = has3 ? __expf(x3 - mx) : 0.f;
            float sum = (e0 + e1) + (e2 + e3);
            #pragma unroll
            for (int off = 16; off >= 1; off >>= 1)
                sum += __shfl_xor(sum, off, 32);
            float inv = 1.0f / sum;
            sS[base + lane]      = e0 * inv;
            sS[base + lane + 32] = e1 * inv;
            sS[base + lane + 64] = e2 * inv;
            if (has3) sS[base + lane + 96] = e3 * inv;
        }
        __syncthreads();

        // ---- 4) out_h = P @ V, chained K=4 WMMAs (C-accumulate) ----
        if (wid < NT) {                // wave-uniform branch: EXEC stays full
            int mt = wid;
            int m  = mt * 16 + l16;
            v8f acc = {};
            for (int kt = 0; kt < 26; ++kt) {   // 104 >= 102; V rows padded 0
                int ks = kt * 4 + 2 * half;
                v2f a, bm;
                a.x  = sS[m * SROW + ks];
                a.y  = sS[m * SROW + ks + 1];
                bm.x = (l16 < HDIM) ? hV[ks * HDIM + l16] : 0.f;
                bm.y = (l16 < HDIM) ? hV[(ks + 1) * HDIM + l16] : 0.f;
                acc = __builtin_amdgcn_wmma_f32_16x16x4_f32(
                          false, a, false, bm, (short)0, acc, false, false);
            }
            if (l16 < HDIM) {
                #pragma unroll
                for (int r = 0; r < 8; ++r) {
                    int M = mt * 16 + r + 8 * half;
                    if (M < SEQ) sO[M * HID + h * HDIM + l16] = acc[r];
                }
            }
        }
        __syncthreads();  // sS free for next head; sO complete after h=1
    }

    // ---- 5) final projection: out = concat_heads @ wo^T, float4 stores ----
    float* op = out + (size_t)b * SEQ * HID;
    for (int idx = tid; idx < SEQ * 2; idx += 256) {
        int s  = idx >> 1;
        int j0 = (idx & 1) * 4;
        float o[4];
        #pragma unroll
        for (int jj = 0; jj < 4; ++jj) {
            float acc = 0.f;
            #pragma unroll
            for (int i = 0; i < HID; ++i)
                acc += sO[s * HID + i] * sW[3 * 64 + (j0 + jj) * HID + i];
            o[jj] = acc;
        }
        float4 r4 = { o[0], o[1], o[2], o[3] };
        *(float4*)(op + s * HID + j0) = r4;
    }
}

extern "C" void kernel_launch(void* const* d_in, const int* in_sizes, int n_in,
                              void* d_out, int out_size, void* d_ws, size_t ws_size,
                              hipStream_t stream) {
    const float* q  = (const float*)d_in[0];
    const float* k  = (const float*)d_in[1];
    const float* v  = (const float*)d_in[2];
    const float* wq = (const float*)d_in[3];
    const float* wk = (const float*)d_in[4];
    const float* wv = (const float*)d_in[5];
    const float* wo = (const float*)d_in[6];
    float* o = (float*)d_out;
    int nb = in_sizes[0] / (SEQ * HID);   // = B = 4096
    mha_rope_kernel<<<nb, 256, 0, stream>>>(q, k, v, wq, wk, wv, wo, o);
}